// NeuralPoints_Simple_Incre_71760313582274
// MI455X (gfx1250) — compile-verified
//
#include <hip/hip_runtime.h>
#include <stdint.h>

// ---------------------------------------------------------------------------
// Gather kernel set for the NeuralPoints sampling reference.
//   out layout (floats), N = #samples = B*R*SR*K:
//     [0      , 3N )  sampled_xyz   (N,3)
//     [3N     , 35N)  sampled_emb   (N,32)
//     [35N    , 38N)  sampled_color (N,3)
//     [38N    , 41N)  sampled_dir   (N,3)
//     [41N    , 42N)  sample_pnt_mask as float 0/1
// ---------------------------------------------------------------------------

#define EMBD        32
#define EMB_BLOCK   256                    // 8 waves / block (wave32)
#define WAVES_PER_BLOCK (EMB_BLOCK / 32)
#define EMB_BATCH   8                      // quads in flight per wave (8*512B LDS)

// ---------------------------------------------------------------------------
// Dominant stream: emb rows (128B each), moved via the CDNA5 async
// global->LDS->global path.  One wave handles a "quad" of 4 samples per
// instruction: lane = {sample-in-quad[1:0] = lane>>3, chunk[2:0] = lane&7},
// each lane moves one 16B float4 chunk of a 32-float emb row.
// Output per quad is 512B fully contiguous -> perfectly coalesced stores.
// Outstanding bytes live in LDS (320KB/WGP), tracked by ASYNCcnt, so MLP
// is not limited by VGPR count.
// ---------------------------------------------------------------------------
__global__ __launch_bounds__(EMB_BLOCK)
void emb_gather_async_kernel(const float* __restrict__ points_emb,
                             const int*   __restrict__ sample_pidx,
                             float*       __restrict__ out_emb,
                             int nQuads, int nSamples)
{
    __shared__ unsigned char smem[WAVES_PER_BLOCK * EMB_BATCH * 512];

    const int lane      = threadIdx.x & 31;
    const int waveInBlk = threadIdx.x >> 5;
    const int gwave     = blockIdx.x * WAVES_PER_BLOCK + waveInBlk;
    const int nWaves    = gridDim.x * WAVES_PER_BLOCK;

    const int sub   = lane >> 3;   // sample within quad: 0..3
    const int chunk = lane & 7;    // float4 chunk within emb row: 0..7

    // LDS byte offset of this lane's 16B slot in batch buffer 0.
    // (low 32 bits of a generic pointer to __shared__ == LDS byte address)
    const uint32_t ldsLane =
        (uint32_t)(uintptr_t)(&smem[waveInBlk * EMB_BATCH * 512]) +
        (uint32_t)(lane * 16);

    for (int q0 = gwave; q0 < nQuads; q0 += nWaves * EMB_BATCH) {
        // ---- phase 1: gather-load EMB rows into LDS (async, ASYNCcnt) ----
#pragma unroll
        for (int b = 0; b < EMB_BATCH; ++b) {
            int q = q0 + b * nWaves;
            if (q < nQuads) {
                int s = q * 4 + sub;
                if (s < nSamples) {
                    int raw = sample_pidx[s];
                    int idx = raw < 0 ? 0 : raw;     // reference clamps, no zero-fill
                    const float* src = points_emb + (size_t)idx * EMBD + chunk * 4;
                    uint32_t lds = ldsLane + (uint32_t)(b * 512);
                    asm volatile("global_load_async_to_lds_b128 %0, %1, off"
                                 :: "v"(lds), "v"(src) : "memory");
                }
            }
        }
        asm volatile("s_wait_asynccnt 0x0" ::: "memory");

        // ---- phase 2: stream LDS tiles to the output (async, coalesced,
        //               non-temporal: 168MB write-once must not evict the
        //               L2-resident gather tables) ----
#pragma unroll
        for (int b = 0; b < EMB_BATCH; ++b) {
            int q = q0 + b * nWaves;
            if (q < nQuads) {
                int s = q * 4 + sub;
                if (s < nSamples) {
                    float* dst = out_emb + (size_t)q * 128 + (size_t)(lane * 4);
                    uint32_t lds = ldsLane + (uint32_t)(b * 512);
                    asm volatile("global_store_async_from_lds_b128 %0, %1, off th:TH_STORE_NT"
                                 :: "v"(dst), "v"(lds) : "memory");
                }
            }
        }
        // LDS buffers are reused next iteration -> drain before overwrite.
        asm volatile("s_wait_asynccnt 0x0" ::: "memory");
    }
}

// ---------------------------------------------------------------------------
// Small fields: xyz / color / dir (3 floats each) + mask. One thread per
// sample. Table reads stay RT (tables are L2-resident: 82MB << 192MB L2);
// outputs are write-once -> non-temporal stores so they don't evict tables.
// ---------------------------------------------------------------------------
__global__ __launch_bounds__(256)
void small_fields_kernel(const float* __restrict__ points_pos,
                         const float* __restrict__ points_color,
                         const float* __restrict__ points_dir,
                         const int*   __restrict__ sample_pidx,
                         float* __restrict__ out_xyz,
                         float* __restrict__ out_color,
                         float* __restrict__ out_dir,
                         float* __restrict__ out_mask,
                         int n)
{
    int s = blockIdx.x * blockDim.x + threadIdx.x;
    if (s >= n) return;

    // stream-ahead prefetch of the index array (emits global_prefetch_b8)
    if (s + 16384 < n) __builtin_prefetch(&sample_pidx[s + 16384], 0, 0);

    int   raw = sample_pidx[s];
    float m   = (raw >= 0) ? 1.0f : 0.0f;
    size_t r  = (size_t)(raw < 0 ? 0 : raw) * 3;
    size_t o  = (size_t)s * 3;

    float p0 = points_pos[r + 0], p1 = points_pos[r + 1], p2 = points_pos[r + 2];
    float c0 = points_color[r + 0], c1 = points_color[r + 1], c2 = points_color[r + 2];
    float d0 = points_dir[r + 0], d1 = points_dir[r + 1], d2 = points_dir[r + 2];

    __builtin_nontemporal_store(p0, out_xyz + o + 0);
    __builtin_nontemporal_store(p1, out_xyz + o + 1);
    __builtin_nontemporal_store(p2, out_xyz + o + 2);
    __builtin_nontemporal_store(c0, out_color + o + 0);
    __builtin_nontemporal_store(c1, out_color + o + 1);
    __builtin_nontemporal_store(c2, out_color + o + 2);
    __builtin_nontemporal_store(d0, out_dir + o + 0);
    __builtin_nontemporal_store(d1, out_dir + o + 1);
    __builtin_nontemporal_store(d2, out_dir + o + 2);
    __builtin_nontemporal_store(m,  out_mask + s);
}

// ---------------------------------------------------------------------------
extern "C" void kernel_launch(void* const* d_in, const int* in_sizes, int n_in,
                              void* d_out, int out_size, void* d_ws, size_t ws_size,
                              hipStream_t stream)
{
    const float* points_pos   = (const float*)d_in[0];
    const float* points_emb   = (const float*)d_in[1];
    const float* points_color = (const float*)d_in[2];
    const float* points_dir   = (const float*)d_in[3];
    const int*   sample_pidx  = (const int*)d_in[4];

    const int N = in_sizes[4];          // total #samples = B*R*SR*K

    float* out       = (float*)d_out;
    float* out_xyz   = out;
    float* out_emb   = out + (size_t)3  * N;
    float* out_color = out + (size_t)35 * N;
    float* out_dir   = out + (size_t)38 * N;
    float* out_mask  = out + (size_t)41 * N;

    const int nQuads = (N + 3) / 4;

    int embGrid = (nQuads + WAVES_PER_BLOCK - 1) / WAVES_PER_BLOCK;
    if (embGrid > 1024) embGrid = 1024;   // grid-stride loop covers the rest
    if (embGrid < 1)    embGrid = 1;

    emb_gather_async_kernel<<<embGrid, EMB_BLOCK, 0, stream>>>(
        points_emb, sample_pidx, out_emb, nQuads, N);

    small_fields_kernel<<<(N + 255) / 256, 256, 0, stream>>>(
        points_pos, points_color, points_dir, sample_pidx,
        out_xyz, out_color, out_dir, out_mask, N);
}